// GraphSAGE_GAT_73538430042648
// MI455X (gfx1250) — compile-verified
//
#include <hip/hip_runtime.h>

// Problem constants (match reference).
#define NN  100000      // num nodes
#define NE  1600000     // num edges
#define DD  32          // hidden dim
#define NH  2           // GAT heads
#define NEF 16          // edge feature dim

typedef __attribute__((ext_vector_type(2))) float    v2f;
typedef __attribute__((ext_vector_type(8))) float    v8f;
typedef __attribute__((ext_vector_type(4))) unsigned ui32x4;
typedef __attribute__((ext_vector_type(8))) int      i32x8;
typedef __attribute__((ext_vector_type(4))) int      i32x4;

// D = A(16x4,f32) * B(4x16,f32) + C(16x16,f32)  -- v_wmma_f32_16x16x4_f32
__device__ __forceinline__ v8f wmma4(v2f a, v2f b, v8f c) {
  return __builtin_amdgcn_wmma_f32_16x16x4_f32(false, a, false, b, (short)0, c,
                                               false, false);
}

// Order-preserving float <-> uint encoding for atomic float-max.
__device__ __forceinline__ unsigned enc_f32(float f) {
  unsigned u = __float_as_uint(f);
  return (u & 0x80000000u) ? ~u : (u | 0x80000000u);
}
__device__ __forceinline__ float dec_f32(unsigned u) {
  return (u & 0x80000000u) ? __uint_as_float(u & 0x7FFFFFFFu)
                           : __uint_as_float(~u);
}

// Issue a 1-D TDM copy of `ndw4*4` bytes... (elems f32) global -> LDS.
// D# built per CDNA5 ISA.
__device__ __forceinline__ void tdm_load_1d(unsigned lds_off,
                                            const float* gptr,
                                            unsigned nelem) {
  unsigned long long ga = (unsigned long long)gptr;
  ui32x4 g0;
  g0[0] = 0x1u;                                   // count=1, user mode
  g0[1] = lds_off;                                // lds_addr (bytes)
  g0[2] = (unsigned)(ga & 0xFFFFFFFFull);         // global_addr[31:0]
  g0[3] = (unsigned)((ga >> 32) & 0x1FFFFFFull)   // global_addr[56:32]
        | 0x80000000u;                            // type=2 ("image")
  i32x8 g1;
  g1[0] = (int)(2u << 16);                        // data_size=4B; mask=0
  g1[1] = (int)((nelem & 0xFFFFu) << 16);         // tensor_dim0[15:0]
  g1[2] = (int)((nelem >> 16) & 0xFFFFu)          // tensor_dim0[31:16]
        | (int)(1u << 16);                        // tensor_dim1 = 1
  g1[3] = (int)((nelem & 0xFFFFu) << 16);         // tile_dim0 = nelem
  g1[4] = 0;                                      // tile_dim1/2 unused
  g1[5] = (int)nelem;                             // tensor_dim0_stride lo
  g1[6] = 0;
  g1[7] = 0;
  i32x4 z4 = {0, 0, 0, 0};
#if __clang_major__ >= 23
  i32x8 z8 = {0, 0, 0, 0, 0, 0, 0, 0};
  __builtin_amdgcn_tensor_load_to_lds(g0, g1, z4, z4, z8, 0);
#else
  __builtin_amdgcn_tensor_load_to_lds(g0, g1, z4, z4, 0);
#endif
}

// ---------------- degree / scatter (SAGE mean aggregation) ----------------
__global__ void deg_kernel(const int* __restrict__ dst, float* __restrict__ deg) {
  int e = blockIdx.x * blockDim.x + threadIdx.x;
  if (e < NE) atomicAdd(&deg[dst[e]], 1.0f);
}

__global__ void scatter_kernel(const int* __restrict__ src,
                               const int* __restrict__ dst,
                               const float* __restrict__ xin,
                               float* __restrict__ agg) {
  int t = blockIdx.x * blockDim.x + threadIdx.x;   // NE*DD = 51.2M < 2^31
  if (t >= NE * DD) return;
  int e = t >> 5, d = t & 31;
  atomicAdd(&agg[dst[e] * DD + d], xin[src[e] * DD + d]);
}

// ------------- SAGE linear: relu(mean @ wl + bl + x @ wr) via WMMA --------
__global__ void sage_linear_kernel(const float* __restrict__ xin,
                                   const float* __restrict__ agg,
                                   const float* __restrict__ deg,
                                   const float* __restrict__ wl,
                                   const float* __restrict__ bl,
                                   const float* __restrict__ wr,
                                   float* __restrict__ xout) {
  const int wave = (blockIdx.x * blockDim.x + threadIdx.x) >> 5;
  const int lane = threadIdx.x & 31;
  const int hi = lane >> 4, r = lane & 15;
  const int n0 = wave * 16;
  if (n0 >= NN) return;                       // wave-uniform; EXEC stays full

  const float dinv = 1.0f / fmaxf(deg[n0 + r], 1.0f);

  v8f acc0, acc1;
  const float b0 = bl[r], b1 = bl[16 + r];
#pragma unroll
  for (int i = 0; i < 8; ++i) { acc0[i] = b0; acc1[i] = b1; }

#pragma unroll
  for (int k0 = 0; k0 < DD; k0 += 4) {
    const int kk = k0 + 2 * hi;
    v2f am, ax, b;
    am.x = agg[(n0 + r) * DD + kk + 0] * dinv;
    am.y = agg[(n0 + r) * DD + kk + 1] * dinv;
    ax.x = xin[(n0 + r) * DD + kk + 0];
    ax.y = xin[(n0 + r) * DD + kk + 1];
    b.x = wl[(kk + 0) * DD + r];      b.y = wl[(kk + 1) * DD + r];
    acc0 = wmma4(am, b, acc0);
    b.x = wl[(kk + 0) * DD + 16 + r]; b.y = wl[(kk + 1) * DD + 16 + r];
    acc1 = wmma4(am, b, acc1);
    b.x = wr[(kk + 0) * DD + r];      b.y = wr[(kk + 1) * DD + r];
    acc0 = wmma4(ax, b, acc0);
    b.x = wr[(kk + 0) * DD + 16 + r]; b.y = wr[(kk + 1) * DD + 16 + r];
    acc1 = wmma4(ax, b, acc1);
  }
#pragma unroll
  for (int i = 0; i < 8; ++i) {
    const int row = n0 + i + 8 * hi;
    xout[row * DD + r]      = fmaxf(acc0[i], 0.0f);
    xout[row * DD + 16 + r] = fmaxf(acc1[i], 0.0f);
  }
}

// ------------------- GAT: xh = x @ gat_w (N x 64) via WMMA ----------------
__global__ void gat_transform_kernel(const float* __restrict__ xin,
                                     const float* __restrict__ w,
                                     float* __restrict__ xh) {
  const int wave = (blockIdx.x * blockDim.x + threadIdx.x) >> 5;
  const int lane = threadIdx.x & 31;
  const int hi = lane >> 4, r = lane & 15;
  const int n0 = wave * 16;
  if (n0 >= NN) return;

  v8f acc[4];
#pragma unroll
  for (int t = 0; t < 4; ++t)
#pragma unroll
    for (int i = 0; i < 8; ++i) acc[t][i] = 0.0f;

#pragma unroll
  for (int k0 = 0; k0 < DD; k0 += 4) {
    const int kk = k0 + 2 * hi;
    v2f a;
    a.x = xin[(n0 + r) * DD + kk + 0];
    a.y = xin[(n0 + r) * DD + kk + 1];
#pragma unroll
    for (int t = 0; t < 4; ++t) {
      v2f b;
      b.x = w[(kk + 0) * 64 + t * 16 + r];
      b.y = w[(kk + 1) * 64 + t * 16 + r];
      acc[t] = wmma4(a, b, acc[t]);
    }
  }
#pragma unroll
  for (int t = 0; t < 4; ++t)
#pragma unroll
    for (int i = 0; i < 8; ++i)
      xh[(n0 + i + 8 * hi) * 64 + t * 16 + r] = acc[t][i];
}

// a_s[n,h] = <xh[n,h,:], att_src[h,:]>, same for a_d
__global__ void gat_att_kernel(const float* __restrict__ xh,
                               const float* __restrict__ att_src,
                               const float* __restrict__ att_dst,
                               float* __restrict__ a_s, float* __restrict__ a_d) {
  int t = blockIdx.x * blockDim.x + threadIdx.x;
  if (t >= NN * NH) return;
  int n = t >> 1, h = t & 1;
  float ss = 0.0f, sd = 0.0f;
#pragma unroll
  for (int k = 0; k < DD; ++k) {
    float v = xh[n * 64 + h * DD + k];
    ss += v * att_src[h * DD + k];
    sd += v * att_dst[h * DD + k];
  }
  a_s[t] = ss;
  a_d[t] = sd;
}

__global__ void gat_minit_kernel(unsigned* __restrict__ m_enc) {
  int t = blockIdx.x * blockDim.x + threadIdx.x;
  if (t < NN * NH) m_enc[t] = 0x007FFFFFu;   // enc(-inf)
}

// segment max of leaky_relu(a_s[src]+a_d[dst]) over incoming edges + loops
__global__ void gat_alpha_max_kernel(const int* __restrict__ src,
                                     const int* __restrict__ dst,
                                     const float* __restrict__ a_s,
                                     const float* __restrict__ a_d,
                                     unsigned* __restrict__ m_enc) {
  int t = blockIdx.x * blockDim.x + threadIdx.x;
  if (t >= NE + NN) return;
  int s, d;
  if (t < NE) { s = src[t]; d = dst[t]; } else { s = t - NE; d = s; }
#pragma unroll
  for (int h = 0; h < NH; ++h) {
    float al = a_s[s * NH + h] + a_d[d * NH + h];
    al = (al > 0.0f) ? al : 0.2f * al;
    atomicMax(&m_enc[d * NH + h], enc_f32(al));
  }
}

// accumulate den += e and gacc += xh[src]*e  (one wave per edge, lane = d)
__global__ void gat_acc_kernel(const int* __restrict__ src,
                               const int* __restrict__ dst,
                               const float* __restrict__ a_s,
                               const float* __restrict__ a_d,
                               const unsigned* __restrict__ m_enc,
                               const float* __restrict__ xh,
                               float* __restrict__ den,
                               float* __restrict__ gacc) {
  const int wave = (blockIdx.x * blockDim.x + threadIdx.x) >> 5;
  const int lane = threadIdx.x & 31;
  if (wave >= NE + NN) return;
  int s, d;
  if (wave < NE) { s = src[wave]; d = dst[wave]; } else { s = wave - NE; d = s; }
#pragma unroll
  for (int h = 0; h < NH; ++h) {
    float al = a_s[s * NH + h] + a_d[d * NH + h];
    al = (al > 0.0f) ? al : 0.2f * al;
    float ex = __expf(al - dec_f32(m_enc[d * NH + h]));
    if (lane == 0) atomicAdd(&den[d * NH + h], ex);
    atomicAdd(&gacc[d * 64 + h * DD + lane], xh[s * 64 + h * DD + lane] * ex);
  }
}

// x3 = relu(mean_h(gacc/(den+eps)) + bias)
__global__ void gat_final_kernel(const float* __restrict__ gacc,
                                 const float* __restrict__ den,
                                 const float* __restrict__ bias,
                                 float* __restrict__ x3) {
  int t = blockIdx.x * blockDim.x + threadIdx.x;
  if (t >= NN * DD) return;
  int n = t >> 5, d = t & 31;
  float v0 = gacc[n * 64 + d]      / (den[n * NH + 0] + 1e-16f);
  float v1 = gacc[n * 64 + DD + d] / (den[n * NH + 1] + 1e-16f);
  x3[t] = fmaxf(0.5f * (v0 + v1) + bias[d], 0.0f);
}

// --------- edge MLP: relu([x[s],x[d],ea] @ w1 + b1) @ w2 + b2 (WMMA) ------
// Per wave: 16 edges. A-tile (16x80) staged in LDS via coalesced b128 loads;
// w1 (80x32) staged once per block via TDM tensor_load_to_lds.
#define AROW 84                       // padded row stride (bank-conflict free)
#define ATILE (16 * AROW)             // 1344 floats per wave
#define W1OFF (8 * ATILE)             // w1 LDS region offset (floats)

__global__ void edge_mlp_kernel(const int* __restrict__ src,
                                const int* __restrict__ dst,
                                const float* __restrict__ x,
                                const float* __restrict__ eattr,
                                const float* __restrict__ w1,
                                const float* __restrict__ b1,
                                const float* __restrict__ w2,
                                const float* __restrict__ b2,
                                float* __restrict__ out) {
  __shared__ float smem[W1OFF + 80 * DD];      // 8 A-tiles + w1 = 53 KB
  const int wslot = threadIdx.x >> 5;
  const int wave = (blockIdx.x * blockDim.x + threadIdx.x) >> 5;
  const int lane = threadIdx.x & 31;
  const int hi = lane >> 4, r = lane & 15;
  const int e0 = wave * 16;                    // NE % 128 == 0, grid exact
  float* At  = &smem[wslot * ATILE];
  float* w1s = &smem[W1OFF];

  // Async-stage w1 into LDS with the Tensor Data Mover (one wave per block).
  if (wslot == 0)
    tdm_load_1d((unsigned)(size_t)w1s, w1, 80u * DD);

  // Cooperatively gather 16 edge rows [x[s] | x[d] | eattr] as float4 chunks.
#pragma unroll
  for (int it = 0; it < 10; ++it) {
    int idx = it * 32 + lane;                  // 0..319 = 16 edges x 20 chunks
    int ei = idx / 20;
    int c  = idx % 20;
    int e  = e0 + ei;
    const float4* p;
    if (c < 8)        p = (const float4*)&x[src[e] * DD + c * 4];
    else if (c < 16)  p = (const float4*)&x[dst[e] * DD + (c - 8) * 4];
    else              p = (const float4*)&eattr[e * NEF + (c - 16) * 4];
    *(float4*)&At[ei * AROW + c * 4] = *p;
  }

  if (wslot == 0) __builtin_amdgcn_s_wait_tensorcnt(0);
  __syncthreads();                             // w1s visible to all waves

  v8f acc0, acc1;
  const float bb0 = b1[r], bb1 = b1[16 + r];
#pragma unroll
  for (int i = 0; i < 8; ++i) { acc0[i] = bb0; acc1[i] = bb1; }

#pragma unroll
  for (int k0 = 0; k0 < 2 * DD + NEF; k0 += 4) {   // K = 80
    const int kk = k0 + 2 * hi;
    v2f a = *(const v2f*)&At[r * AROW + kk];       // ds_load_b64, no conflicts
    v2f b;
    b.x = w1s[(kk + 0) * DD + r];      b.y = w1s[(kk + 1) * DD + r];
    acc0 = wmma4(a, b, acc0);
    b.x = w1s[(kk + 0) * DD + 16 + r]; b.y = w1s[(kk + 1) * DD + 16 + r];
    acc1 = wmma4(a, b, acc1);
  }

  // Hidden tile (16x32), reusing this wave's A region (LDS ops are in-order).
#pragma unroll
  for (int i = 0; i < 8; ++i) {
    At[(i + 8 * hi) * 33 + r]      = fmaxf(acc0[i], 0.0f);
    At[(i + 8 * hi) * 33 + 16 + r] = fmaxf(acc1[i], 0.0f);
  }

  // Second layer: 32 -> 1 dot; each half-wave handles 16 columns of row r.
  float part = 0.0f;
#pragma unroll
  for (int c = 0; c < 16; ++c)
    part += At[r * 33 + hi * 16 + c] * w2[hi * 16 + c];
  part += __shfl_xor(part, 16, 32);
  if (hi == 0) out[e0 + r] = part + b2[0];
}

// --------------------------------- launch ---------------------------------
extern "C" void kernel_launch(void* const* d_in, const int* in_sizes, int n_in,
                              void* d_out, int out_size, void* d_ws, size_t ws_size,
                              hipStream_t stream) {
  const int*   edge_index = (const int*)d_in[0];
  const int*   src  = edge_index;
  const int*   dst  = edge_index + NE;
  const float* eattr = (const float*)d_in[1];
  const float* x0   = (const float*)d_in[2];
  const float* s1wl = (const float*)d_in[3];
  const float* s1bl = (const float*)d_in[4];
  const float* s1wr = (const float*)d_in[5];
  const float* s2wl = (const float*)d_in[6];
  const float* s2bl = (const float*)d_in[7];
  const float* s2wr = (const float*)d_in[8];
  const float* gw   = (const float*)d_in[9];
  const float* gas  = (const float*)d_in[10];
  const float* gad  = (const float*)d_in[11];
  const float* gb   = (const float*)d_in[12];
  const float* w1   = (const float*)d_in[13];
  const float* b1   = (const float*)d_in[14];
  const float* w2   = (const float*)d_in[15];
  const float* b2   = (const float*)d_in[16];
  float* out = (float*)d_out;

  // Workspace carve-up (~93 MB); everything read is initialized below.
  float* ws   = (float*)d_ws;
  float* deg  = ws;                 ws += NN;
  float* agg  = ws;                 ws += (size_t)NN * DD;   // contiguous w/ deg
  float* x1   = ws;                 ws += (size_t)NN * DD;
  float* x2   = ws;                 ws += (size_t)NN * DD;
  float* xh   = ws;                 ws += (size_t)NN * 64;
  float* a_s  = ws;                 ws += (size_t)NN * NH;
  float* a_d  = ws;                 ws += (size_t)NN * NH;
  unsigned* m_enc = (unsigned*)ws;  ws += (size_t)NN * NH;
  float* den  = ws;                 ws += (size_t)NN * NH;   // contiguous w/ gacc
  float* gacc = ws;                 ws += (size_t)NN * 64;
  float* x3   = x1;                 // x1 is dead after SAGE layer 2

  const int B = 256;
  const int gEdges   = (NE + B - 1) / B;
  const int gScatter = (NE * DD + B - 1) / B;
  const int gNodeWav = ((NN / 16) * 32 + B - 1) / B;
  const int gNH      = (NN * NH + B - 1) / B;
  const int gND      = (NN * DD + B - 1) / B;
  const int gEdgeN   = (NE + NN + B - 1) / B;
  const int gAccWav  = (int)(((long long)(NE + NN) * 32 + B - 1) / B);
  const int gMlp     = (NE / 16) * 32 / B;   // exact: 12500 blocks

  // ---- SAGE layer 1 ----
  hipMemsetAsync(deg, 0, sizeof(float) * (size_t)NN * (1 + DD), stream); // deg+agg
  deg_kernel<<<gEdges, B, 0, stream>>>(dst, deg);
  scatter_kernel<<<gScatter, B, 0, stream>>>(src, dst, x0, agg);
  sage_linear_kernel<<<gNodeWav, B, 0, stream>>>(x0, agg, deg, s1wl, s1bl, s1wr, x1);

  // ---- SAGE layer 2 ----
  hipMemsetAsync(agg, 0, sizeof(float) * (size_t)NN * DD, stream);
  scatter_kernel<<<gScatter, B, 0, stream>>>(src, dst, x1, agg);
  sage_linear_kernel<<<gNodeWav, B, 0, stream>>>(x1, agg, deg, s2wl, s2bl, s2wr, x2);

  // ---- GAT ----
  gat_transform_kernel<<<gNodeWav, B, 0, stream>>>(x2, gw, xh);
  gat_att_kernel<<<gNH, B, 0, stream>>>(xh, gas, gad, a_s, a_d);
  gat_minit_kernel<<<gNH, B, 0, stream>>>(m_enc);
  hipMemsetAsync(den, 0, sizeof(float) * (size_t)NN * (NH + 64), stream); // den+gacc
  gat_alpha_max_kernel<<<gEdgeN, B, 0, stream>>>(src, dst, a_s, a_d, m_enc);
  gat_acc_kernel<<<gAccWav, B, 0, stream>>>(src, dst, a_s, a_d, m_enc, xh, den, gacc);
  gat_final_kernel<<<gND, B, 0, stream>>>(gacc, den, gb, x3);

  // ---- edge MLP ----
  edge_mlp_kernel<<<gMlp, B, 0, stream>>>(src, dst, x3, eattr, w1, b1, w2, b2, out);
}